// Encoder_15375982920181
// MI455X (gfx1250) — compile-verified
//
#include <hip/hip_runtime.h>
#include <hip/hip_bf16.h>

typedef __attribute__((ext_vector_type(2))) float v2f;
typedef __attribute__((ext_vector_type(8))) float v8f;

#define USER_NUM 50000
#define ITEM_NUM 50000
#define N_NODES  (USER_NUM + ITEM_NUM)
#define N_EDGES  1000000
#define IN_DIM   128
#define HIDE_DIM 64

// ---------------------------------------------------------------------------
// generic zero fill
// ---------------------------------------------------------------------------
__global__ void zero_kernel(float* __restrict__ p, int n) {
    int i = blockIdx.x * blockDim.x + threadIdx.x;
    int stride = gridDim.x * blockDim.x;
    for (; i < n; i += stride) p[i] = 0.0f;
}

// ---------------------------------------------------------------------------
// degree accumulation: deg_out[src]++, deg_in[dst]++
// ---------------------------------------------------------------------------
__global__ void degree_kernel(const int* __restrict__ src, const int* __restrict__ dst,
                              float* __restrict__ deg_out, float* __restrict__ deg_in, int n) {
    int i = blockIdx.x * blockDim.x + threadIdx.x;
    if (i >= n) return;
    atomicAdd(&deg_out[src[i]], 1.0f);
    atomicAdd(&deg_in[dst[i]],  1.0f);
}

// deg -> clip(deg,1)^(-1/2), in place over both arrays (contiguous)
__global__ void finalize_deg_kernel(float* __restrict__ p, int n) {
    int i = blockIdx.x * blockDim.x + threadIdx.x;
    if (i >= n) return;
    float d = p[i];
    d = d < 1.0f ? 1.0f : d;
    p[i] = 1.0f / sqrtf(d);
}

// ---------------------------------------------------------------------------
// WMMA GEMM: out = relu(A[M x 128] @ W[128 x 64] + bias), written to two
// destinations (d_out embedding region and the layer-0 feature buffer).
// 256 threads = 8 waves; each wave computes one 16-row M tile across N=64
// using V_WMMA_F32_16X16X4_F32 (full fp32 precision, 32 K-steps).
// W is staged into LDS with GLOBAL_LOAD_ASYNC_TO_LDS_B128 (ASYNCcnt path).
// ---------------------------------------------------------------------------
__global__ __launch_bounds__(256) void gemm_bias_relu_kernel(
        const float* __restrict__ A, const float* __restrict__ W,
        const float* __restrict__ bias,
        float* __restrict__ out0, float* __restrict__ out1, int M) {
    __shared__ float Wlds[IN_DIM * HIDE_DIM];   // 32 KB
    __shared__ float blds[HIDE_DIM];

    // stage W into LDS via async tensor/LDS DMA path (no VGPR round trip).
    // GVS addressing: mem = SADDR(64b) + VADDR(32b offset); LDS dest = VDST VGPR.
    {
        const unsigned lds_base = (unsigned)(uintptr_t)(&Wlds[0]); // low 32 bits of flat = LDS offset
        #pragma unroll
        for (int i = 0; i < 8; ++i) {
            const unsigned off = (threadIdx.x + i * 256u) * 16u;   // byte offset, 2048 x b128 total
            asm volatile("global_load_async_to_lds_b128 %0, %1, %2"
                         :: "v"(lds_base + off), "v"(off), "s"(W)
                         : "memory");
        }
        if (threadIdx.x < HIDE_DIM) blds[threadIdx.x] = bias[threadIdx.x];
        asm volatile("s_wait_asynccnt 0x0" ::: "memory");
    }
    __syncthreads();

    const int wave = threadIdx.x >> 5;
    const int lane = threadIdx.x & 31;
    const int tile = blockIdx.x * 8 + wave;         // 16-row M tile index
    if (tile * 16 >= M) return;                     // wave-uniform: EXEC stays all-ones

    const int rc  = lane & 15;                      // A row within tile == C/B column within N-tile
    const int kh  = (lane >> 4) << 1;               // K sub-offset: 0 (lanes 0-15) or 2 (lanes 16-31)
    const float* __restrict__ Arow = A + (size_t)(tile * 16 + rc) * IN_DIM;

    v8f acc0 = {}, acc1 = {}, acc2 = {}, acc3 = {};

    #pragma unroll 4
    for (int k0 = 0; k0 < IN_DIM; k0 += 4) {
        v2f a;
        a.x = Arow[k0 + kh + 0];
        a.y = Arow[k0 + kh + 1];
        const float* __restrict__ w0 = &Wlds[(k0 + kh + 0) * HIDE_DIM + rc];
        const float* __restrict__ w1 = &Wlds[(k0 + kh + 1) * HIDE_DIM + rc];
        v2f b0; b0.x = w0[ 0]; b0.y = w1[ 0];
        v2f b1; b1.x = w0[16]; b1.y = w1[16];
        v2f b2; b2.x = w0[32]; b2.y = w1[32];
        v2f b3; b3.x = w0[48]; b3.y = w1[48];
        acc0 = __builtin_amdgcn_wmma_f32_16x16x4_f32(false, a, false, b0, (short)0, acc0, false, false);
        acc1 = __builtin_amdgcn_wmma_f32_16x16x4_f32(false, a, false, b1, (short)0, acc1, false, false);
        acc2 = __builtin_amdgcn_wmma_f32_16x16x4_f32(false, a, false, b2, (short)0, acc2, false, false);
        acc3 = __builtin_amdgcn_wmma_f32_16x16x4_f32(false, a, false, b3, (short)0, acc3, false, false);
    }

    // C layout: lane L holds col (L%16 + 16*t); VGPR r -> row r + 8*(L/16)
    const int rbase = (lane >> 4) << 3;
    v8f accs[4] = {acc0, acc1, acc2, acc3};
    #pragma unroll
    for (int t = 0; t < 4; ++t) {
        const int col = t * 16 + rc;
        const float bv = blds[col];
        #pragma unroll
        for (int r = 0; r < 8; ++r) {
            const int m = tile * 16 + rbase + r;
            float v = accs[t][r] + bv;
            v = v > 0.0f ? v : 0.0f;
            const size_t idx = (size_t)m * HIDE_DIM + col;
            out0[idx] = v;
            out1[idx] = v;
        }
    }
}

// ---------------------------------------------------------------------------
// edge scatter: agg[dst] += feat[src] * norm_out[src]   (64 threads per edge)
// ---------------------------------------------------------------------------
__global__ void scatter_kernel(const int* __restrict__ src, const int* __restrict__ dst,
                               const float* __restrict__ feat, const float* __restrict__ norm_out,
                               float* __restrict__ agg, int n_edges) {
    int tid = blockIdx.x * blockDim.x + threadIdx.x;
    int e = tid >> 6;
    int d = tid & 63;
    if (e >= n_edges) return;
    int s = src[e];
    int t = dst[e];
    float v = feat[(size_t)s * HIDE_DIM + d] * norm_out[s];
    atomicAdd(&agg[(size_t)t * HIDE_DIM + d], v);
}

// ---------------------------------------------------------------------------
// layer update: a = agg*norm_in; emb += a*scale; feat_next = a
// ---------------------------------------------------------------------------
__global__ void update_kernel(const float* __restrict__ agg, const float* __restrict__ norm_in,
                              float* __restrict__ emb, float* __restrict__ feat,
                              float scale, int n_nodes) {
    int tid = blockIdx.x * blockDim.x + threadIdx.x;
    int n = tid >> 6;
    if (n >= n_nodes) return;
    float a = agg[tid] * norm_in[n];
    emb[tid] += a * scale;
    feat[tid] = a;
}

// ---------------------------------------------------------------------------
// item_emb += item_side_feat
// ---------------------------------------------------------------------------
__global__ void add_side_kernel(float* __restrict__ item_emb, const float* __restrict__ side, int n) {
    int i = blockIdx.x * blockDim.x + threadIdx.x;
    if (i >= n) return;
    item_emb[i] += side[i];
}

extern "C" void kernel_launch(void* const* d_in, const int* in_sizes, int n_in,
                              void* d_out, int out_size, void* d_ws, size_t ws_size,
                              hipStream_t stream) {
    const float* user_in  = (const float*)d_in[0];
    const float* item_in  = (const float*)d_in[1];
    const float* side     = (const float*)d_in[2];
    const int*   src      = (const int*)d_in[3];
    const int*   dst      = (const int*)d_in[4];
    const float* Wu       = (const float*)d_in[5];
    const float* bu       = (const float*)d_in[6];
    const float* Wi       = (const float*)d_in[7];
    const float* bi       = (const float*)d_in[8];
    float* out = (float*)d_out;                // [user_emb (50000x64) | item_emb (50000x64)]

    const int n_edges = in_sizes[3];

    // workspace layout (floats)
    float* ws       = (float*)d_ws;
    float* norm_out = ws;                                   // N_NODES
    float* norm_in  = ws + N_NODES;                         // N_NODES
    float* feat     = ws + 2 * (size_t)N_NODES;             // N_NODES * 64
    float* agg      = feat + (size_t)N_NODES * HIDE_DIM;    // N_NODES * 64

    // 1) degrees -> norms
    zero_kernel<<<512, 256, 0, stream>>>(norm_out, 2 * N_NODES);
    degree_kernel<<<(n_edges + 255) / 256, 256, 0, stream>>>(src, dst, norm_out, norm_in, n_edges);
    finalize_deg_kernel<<<(2 * N_NODES + 255) / 256, 256, 0, stream>>>(norm_out, 2 * N_NODES);

    // 2) input projections (WMMA GEMM + bias + relu); also initializes layer-0 features
    const int gemm_blocks = (USER_NUM / 16 + 7) / 8;  // 3125 tiles / 8 waves
    gemm_bias_relu_kernel<<<gemm_blocks, 256, 0, stream>>>(
        user_in, Wu, bu, out, feat, USER_NUM);
    gemm_bias_relu_kernel<<<gemm_blocks, 256, 0, stream>>>(
        item_in, Wi, bi, out + (size_t)USER_NUM * HIDE_DIM,
        feat + (size_t)USER_NUM * HIDE_DIM, ITEM_NUM);

    // 3) two propagation layers
    const int feat_elems = N_NODES * HIDE_DIM;
    for (int layer = 0; layer < 2; ++layer) {
        zero_kernel<<<2048, 256, 0, stream>>>(agg, feat_elems);
        const long long work = (long long)n_edges * HIDE_DIM;
        scatter_kernel<<<(int)((work + 255) / 256), 256, 0, stream>>>(
            src, dst, feat, norm_out, agg, n_edges);
        const float scale = 1.0f / (float)(layer + 2);
        update_kernel<<<(feat_elems + 255) / 256, 256, 0, stream>>>(
            agg, norm_in, out, feat, scale, N_NODES);
    }

    // 4) item side features
    add_side_kernel<<<(USER_NUM * HIDE_DIM + 255) / 256, 256, 0, stream>>>(
        out + (size_t)USER_NUM * HIDE_DIM, side, ITEM_NUM * HIDE_DIM);
}